// URNNCell_65893388255691
// MI455X (gfx1250) — compile-verified
//
#include <hip/hip_runtime.h>
#include <math.h>
#include <stdint.h>

typedef __attribute__((ext_vector_type(2))) float v2f;
typedef __attribute__((ext_vector_type(4))) float v4f;
typedef __attribute__((ext_vector_type(8))) float v8f;

// ---------------- GEMM: C[M,N] = A[M,K] * Bw[N,K]^T  (all f32, row-major) ----
#define BM 128
#define BN 64
#define BK 32
#define LDS_STRIDE 36   // 32 + 4 pad: distinct banks for 16-lane row-strided reads

// CDNA5 async global->LDS copy (ASYNCcnt-tracked, no VGPR round trip).
// vdst = per-lane LDS byte offset, vaddr = per-lane 64-bit global address.
__device__ __forceinline__ void async_g2l_b128(uint32_t lds_off, const void* gptr) {
  asm volatile("global_load_async_to_lds_b128 %0, %1, off"
               :
               : "v"(lds_off), "v"((uint64_t)(uintptr_t)gptr)
               : "memory");
}
#define WAIT_ASYNCCNT(n) asm volatile("s_wait_asynccnt %0" ::"i"(n) : "memory")

__global__ __launch_bounds__(256)
void gemm_wmma_f32(const float* __restrict__ A, const float* __restrict__ Bw,
                   float* __restrict__ C, int M, int N, int K) {
  __shared__ __align__(16) float As[2][BM * LDS_STRIDE];
  __shared__ __align__(16) float Bs[2][BN * LDS_STRIDE];

  const int t     = threadIdx.x;
  const int lane  = t & 31;
  const int wave  = t >> 5;       // 0..7
  const int wm    = wave >> 1;    // 0..3 : wave row within block
  const int wn    = wave & 1;     // 0..1 : wave col within block
  const int laneM = lane & 15;
  const int kSel  = (lane >> 4) << 1;   // lanes 0-15 -> K+0/K+1, lanes 16-31 -> K+2/K+3

  const int blockN = blockIdx.x * BN;
  const int blockM = blockIdx.y * BM;

  // LDS byte bases of the double buffers (low 32 bits of a generic LDS
  // pointer are the byte offset within the workgroup's LDS allocation).
  const uint32_t asBase = (uint32_t)(uintptr_t)&As[0][0];
  const uint32_t bsBase = (uint32_t)(uintptr_t)&Bs[0][0];
  const uint32_t aBufBytes = (uint32_t)(BM * LDS_STRIDE * sizeof(float));
  const uint32_t bBufBytes = (uint32_t)(BN * LDS_STRIDE * sizeof(float));

  // This thread's fixed tile slots (row / 4-float column group)
  // A tile: 128x32 floats = 1024 float4, 4 per thread. B tile: 512 float4, 2 per thread.
  int aRow[4], aC4[4], bRow[2], bC4[2];
#pragma unroll
  for (int i = 0; i < 4; ++i) {
    int idx = t + i * 256;
    aRow[i] = idx >> 3;
    aC4[i]  = (idx & 7) << 2;
  }
#pragma unroll
  for (int i = 0; i < 2; ++i) {
    int idx = t + i * 256;
    bRow[i] = idx >> 3;
    bC4[i]  = (idx & 7) << 2;
  }

  auto issue_tile = [&](int kk, int buf) {
#pragma unroll
    for (int i = 0; i < 4; ++i) {
      const void* g = A + (size_t)(blockM + aRow[i]) * K + kk + aC4[i];
      uint32_t l = asBase + (uint32_t)buf * aBufBytes +
                   (uint32_t)(aRow[i] * LDS_STRIDE + aC4[i]) * 4u;
      async_g2l_b128(l, g);
    }
#pragma unroll
    for (int i = 0; i < 2; ++i) {
      const void* g = Bw + (size_t)(blockN + bRow[i]) * K + kk + bC4[i];
      uint32_t l = bsBase + (uint32_t)buf * bBufBytes +
                   (uint32_t)(bRow[i] * LDS_STRIDE + bC4[i]) * 4u;
      async_g2l_b128(l, g);
    }
  };

  v8f acc[2][2] = {};

  const int NT = K / BK;
  issue_tile(0, 0);                      // prologue: stage tile 0

  for (int kt = 0; kt < NT; ++kt) {
    const int buf = kt & 1;
    if (kt + 1 < NT) {
      issue_tile((kt + 1) * BK, buf ^ 1);  // overlap next tile's DMA with compute
      WAIT_ASYNCCNT(6);                    // tile kt resident (6 newer still in flight)
    } else {
      WAIT_ASYNCCNT(0);
    }
    __syncthreads();

    const float* as = &As[buf][0];
    const float* bs = &Bs[buf][0];
#pragma unroll
    for (int k = 0; k < BK; k += 4) {
      v2f afrag[2], bfrag[2];
#pragma unroll
      for (int i = 0; i < 2; ++i) {
        int r = wm * 32 + i * 16 + laneM;
        afrag[i] = *(const v2f*)&as[r * LDS_STRIDE + k + kSel];
      }
#pragma unroll
      for (int j = 0; j < 2; ++j) {
        int r = wn * 32 + j * 16 + laneM;
        bfrag[j] = *(const v2f*)&bs[r * LDS_STRIDE + k + kSel];
      }
#pragma unroll
      for (int i = 0; i < 2; ++i)
#pragma unroll
        for (int j = 0; j < 2; ++j)
          acc[i][j] = __builtin_amdgcn_wmma_f32_16x16x4_f32(
              false, afrag[i], false, bfrag[j], (short)0, acc[i][j],
              false, false);
    }
    __syncthreads();   // all reads of this buffer done before it is re-staged
  }

  // C/D layout: VGPR r -> M = r (lanes 0-15) / r+8 (lanes 16-31), N = lane%16
  const int mBase = blockM + wm * 32 + ((lane >> 4) << 3);
  const int nBase = blockN + wn * 32 + laneM;
#pragma unroll
  for (int i = 0; i < 2; ++i)
#pragma unroll
    for (int j = 0; j < 2; ++j)
#pragma unroll
      for (int r = 0; r < 8; ++r)
        C[(size_t)(mBase + i * 16 + r) * N + (nBase + j * 16)] = acc[i][j][r];
}

// ---------------- Per-row URNN pipeline --------------------------------------
#define FFT_N 2048
#define LOG2N 11
#define T2 256
#define PER_T (FFT_N / T2)   // 8
#define URNN_EPS 1e-6f

__device__ __forceinline__ void reflect_lds(float* zr, float* zi,
                                            float* redA, float* redB, float* redC,
                                            const float* __restrict__ vre,
                                            const float* __restrict__ vim,
                                            float preScale, int t) {
  float lr[PER_T], li[PER_T], vr[PER_T], vi[PER_T];
  float svr = 0.f, svi = 0.f, sn = 0.f;
#pragma unroll
  for (int i = 0; i < PER_T; ++i) {
    int n = t + i * T2;
    vr[i] = vre[n]; vi[i] = vim[n];
    float a = zr[n] * preScale;
    float b = zi[n] * preScale;
    lr[i] = a; li[i] = b;
    svr += a * vr[i] + b * vi[i];   // Re(z * conj(v))
    svi += b * vr[i] - a * vi[i];   // Im(z * conj(v))
    sn  += vr[i] * vr[i] + vi[i] * vi[i];
  }
  redA[t] = svr; redB[t] = svi; redC[t] = sn;
  __syncthreads();
  for (int off = T2 / 2; off > 0; off >>= 1) {
    if (t < off) {
      redA[t] += redA[t + off];
      redB[t] += redB[t + off];
      redC[t] += redC[t + off];
    }
    __syncthreads();
  }
  float f   = 2.0f / redC[0];
  float vzr = redA[0] * f;
  float vzi = redB[0] * f;
#pragma unroll
  for (int i = 0; i < PER_T; ++i) {
    int n = t + i * T2;
    zr[n] = lr[i] - (vzr * vr[i] - vzi * vi[i]);
    zi[n] = li[i] - (vzr * vi[i] + vzi * vr[i]);
  }
  __syncthreads();
}

__global__ __launch_bounds__(256)
void urnn_row_kernel(const float* __restrict__ state,
                     const float* __restrict__ inputs_mul,
                     const float* __restrict__ b_h,
                     const float* __restrict__ d1_w,
                     const float* __restrict__ r1_re, const float* __restrict__ r1_im,
                     const float* __restrict__ d2_w,
                     const float* __restrict__ r2_re, const float* __restrict__ r2_im,
                     const float* __restrict__ d3_w,
                     const int*   __restrict__ perm,
                     float* __restrict__ out) {
  __shared__ float zr[FFT_N], zi[FFT_N];
  __shared__ float twr[FFT_N / 2], twi[FFT_N / 2];
  __shared__ float redA[T2], redB[T2], redC[T2];

  const int t = threadIdx.x;
  const int row = blockIdx.x;
  const size_t rbase = (size_t)row * (2 * FFT_N);

  // Twiddle table: W_k = exp(-2*pi*i*k/N), k = 0..N/2-1
  for (int k = t; k < FFT_N / 2; k += T2) {
    float s, c;
    __sincosf(-6.283185307179586f * (float)k / (float)FFT_N, &s, &c);
    twr[k] = c; twi[k] = s;
  }
  // Load state row, apply diag(e^{i d1}), store bit-reversed for DIT FFT
#pragma unroll
  for (int i = 0; i < PER_T; ++i) {
    int n = t + i * T2;
    float sr = state[rbase + n];
    float si = state[rbase + FFT_N + n];
    float c1, s1; __sincosf(d1_w[n], &s1, &c1);
    float a = c1 * sr - s1 * si;
    float b = s1 * sr + c1 * si;
    int rI = (int)(__brev((unsigned)n) >> (32 - LOG2N));
    zr[rI] = a; zi[rI] = b;
  }
  __syncthreads();

  // Forward FFT (radix-2 DIT)
  for (int s = 1; s <= LOG2N; ++s) {
    int half = 1 << (s - 1);
    int tws  = 1 << (LOG2N - s);
#pragma unroll 1
    for (int j = t; j < FFT_N / 2; j += T2) {
      int pos = j & (half - 1);
      int i0  = ((j >> (s - 1)) << s) + pos;
      int i1  = i0 + half;
      float wr = twr[pos * tws], wim = twi[pos * tws];
      float ur = zr[i0], ui = zi[i0];
      float vr = zr[i1], vi = zi[i1];
      float tr = wr * vr - wim * vi;
      float ti = wr * vi + wim * vr;
      zr[i0] = ur + tr; zi[i0] = ui + ti;
      zr[i1] = ur - tr; zi[i1] = ui - ti;
    }
    __syncthreads();
  }

  // Householder reflection #1
  reflect_lds(zr, zi, redA, redB, redC, r1_re, r1_im, 1.0f, t);

  // Permutation gather + diag(e^{i d2}) + bit-reversed store (for IFFT)
  {
    float gr[PER_T], gi[PER_T];
#pragma unroll
    for (int i = 0; i < PER_T; ++i) {
      int n = t + i * T2;
      int p = perm[n];
      gr[i] = zr[p]; gi[i] = zi[p];
    }
    __syncthreads();
#pragma unroll
    for (int i = 0; i < PER_T; ++i) {
      int n = t + i * T2;
      float c2, s2; __sincosf(d2_w[n], &s2, &c2);
      float a = c2 * gr[i] - s2 * gi[i];
      float b = s2 * gr[i] + c2 * gi[i];
      int rI = (int)(__brev((unsigned)n) >> (32 - LOG2N));
      zr[rI] = a; zi[rI] = b;
    }
    __syncthreads();
  }

  // Inverse FFT: conjugate twiddles; 1/N folded into reflection #2 preScale
  for (int s = 1; s <= LOG2N; ++s) {
    int half = 1 << (s - 1);
    int tws  = 1 << (LOG2N - s);
#pragma unroll 1
    for (int j = t; j < FFT_N / 2; j += T2) {
      int pos = j & (half - 1);
      int i0  = ((j >> (s - 1)) << s) + pos;
      int i1  = i0 + half;
      float wr = twr[pos * tws], wim = -twi[pos * tws];
      float ur = zr[i0], ui = zi[i0];
      float vr = zr[i1], vi = zi[i1];
      float tr = wr * vr - wim * vi;
      float ti = wr * vi + wim * vr;
      zr[i0] = ur + tr; zi[i0] = ui + ti;
      zr[i1] = ur - tr; zi[i1] = ui - ti;
    }
    __syncthreads();
  }

  // Householder reflection #2 (with IFFT 1/N scale)
  reflect_lds(zr, zi, redA, redB, redC, r2_re, r2_im, 1.0f / (float)FFT_N, t);

  // diag(e^{i d3}) + add inputs_c + modReLU + store
#pragma unroll
  for (int i = 0; i < PER_T; ++i) {
    int n = t + i * T2;
    float c3, s3; __sincosf(d3_w[n], &s3, &c3);
    float a = zr[n], b = zi[n];
    float wrv = c3 * a - s3 * b;
    float wiv = s3 * a + c3 * b;
    float pr = inputs_mul[rbase + n] + wrv;
    float pi = inputs_mul[rbase + FFT_N + n] + wiv;
    float norm  = sqrtf(pr * pr + pi * pi);
    float scale = fmaxf(norm + b_h[n], 0.0f) / (norm + URNN_EPS);
    out[rbase + n]         = pr * scale;
    out[rbase + FFT_N + n] = pi * scale;
  }
}

// ---------------- Launch -----------------------------------------------------
extern "C" void kernel_launch(void* const* d_in, const int* in_sizes, int n_in,
                              void* d_out, int out_size, void* d_ws, size_t ws_size,
                              hipStream_t stream) {
  const float* inputs = (const float*)d_in[0];
  const float* state  = (const float*)d_in[1];
  const float* w_ih   = (const float*)d_in[2];
  const float* b_h    = (const float*)d_in[3];
  const float* d1_w   = (const float*)d_in[4];
  const float* r1_re  = (const float*)d_in[5];
  const float* r1_im  = (const float*)d_in[6];
  const float* d2_w   = (const float*)d_in[7];
  const float* r2_re  = (const float*)d_in[8];
  const float* r2_im  = (const float*)d_in[9];
  const float* d3_w   = (const float*)d_in[10];
  const int*   perm   = (const int*)d_in[11];
  float* out    = (float*)d_out;
  float* mulbuf = (float*)d_ws;   // inputs @ w_ih.T : B x 2N f32 (128 MB)

  const int N  = in_sizes[3];                 // 2048
  const int B  = in_sizes[1] / (2 * N);       // 8192
  const int K  = in_sizes[0] / B;             // 1024
  const int N2 = 2 * N;                       // 4096

  dim3 g1(N2 / BN, B / BM);
  gemm_wmma_f32<<<g1, 256, 0, stream>>>(inputs, w_ih, mulbuf, B, N2, K);

  urnn_row_kernel<<<dim3(B), 256, 0, stream>>>(
      state, mulbuf, b_h, d1_w, r1_re, r1_im, d2_w, r2_re, r2_im, d3_w, perm,
      out);
}